// VQVAE_26671746908683
// MI455X (gfx1250) — compile-verified
//
#include <hip/hip_runtime.h>
#include <math.h>

typedef __attribute__((ext_vector_type(2))) float v2f;
typedef __attribute__((ext_vector_type(8))) float v8f;

#define EPS   1e-5f
#define SLOPE 0.1f

// ---------------------------------------------------------------- utilities
__global__ void k_zero(float* __restrict__ p, int n) {
  int i = blockIdx.x * blockDim.x + threadIdx.x;
  if (i < n) p[i] = 0.0f;
}

// ---------------------------------------------------- direct 3x3 conv, pad=1
// act: 0 = none, 1 = tanh
__global__ void k_conv2d(const float* __restrict__ in, const float* __restrict__ w,
                         const float* __restrict__ bias, float* __restrict__ out,
                         int N, int Cin, int Hin, int Win,
                         int Cout, int Hout, int Wout, int stride, int act) {
  long i = (long)blockIdx.x * blockDim.x + threadIdx.x;
  long total = (long)N * Cout * Hout * Wout;
  if (i >= total) return;
  int ow = (int)(i % Wout); long t = i / Wout;
  int oh = (int)(t % Hout); t /= Hout;
  int co = (int)(t % Cout); int n = (int)(t / Cout);

  float acc = bias[co];
  for (int ci = 0; ci < Cin; ++ci) {
    const float* ip = in + ((long)(n * Cin + ci)) * Hin * Win;
    const float* wp = w + ((co * Cin + ci) * 9);
    #pragma unroll
    for (int kh = 0; kh < 3; ++kh) {
      int ih = oh * stride + kh - 1;
      if (ih < 0 || ih >= Hin) continue;
      #pragma unroll
      for (int kw = 0; kw < 3; ++kw) {
        int iw = ow * stride + kw - 1;
        if (iw < 0 || iw >= Win) continue;
        acc = fmaf(ip[(long)ih * Win + iw], wp[kh * 3 + kw], acc);
      }
    }
  }
  if (act == 1) acc = tanhf(acc);
  out[i] = acc;
}

// ------------------------- ConvTranspose2d: k=3, stride=2, pad=1, out_pad=1
// weight layout (Cin, Cout, 3, 3); out[oh] += x[ih]*w[kh] where oh = 2*ih + kh - 1
__global__ void k_convT(const float* __restrict__ in, const float* __restrict__ w,
                        const float* __restrict__ bias, float* __restrict__ out,
                        int N, int Cin, int Hin, int Win, int Cout) {
  int Hout = 2 * Hin, Wout = 2 * Win;
  long i = (long)blockIdx.x * blockDim.x + threadIdx.x;
  long total = (long)N * Cout * Hout * Wout;
  if (i >= total) return;
  int ow = (int)(i % Wout); long t = i / Wout;
  int oh = (int)(t % Hout); t /= Hout;
  int co = (int)(t % Cout); int n = (int)(t / Cout);

  float acc = bias[co];
  #pragma unroll
  for (int kh = 0; kh < 3; ++kh) {
    int th = oh + 1 - kh;
    if (th & 1) continue;
    int ih = th >> 1;
    if (ih < 0 || ih >= Hin) continue;
    #pragma unroll
    for (int kw = 0; kw < 3; ++kw) {
      int tw = ow + 1 - kw;
      if (tw & 1) continue;
      int iw = tw >> 1;
      if (iw < 0 || iw >= Win) continue;
      for (int ci = 0; ci < Cin; ++ci) {
        acc = fmaf(in[((long)(n * Cin + ci) * Hin + ih) * Win + iw],
                   w[((ci * Cout + co) * 3 + kh) * 3 + kw], acc);
      }
    }
  }
  out[i] = acc;
}

// ------------------------------------- BatchNorm (training) stats: sum/sumsq
// One 256-thread block covers a contiguous chunk of one (n,c) plane (HW%256==0).
__global__ void k_bn_stats(const float* __restrict__ x, float* __restrict__ stats,
                           int C, int HW) {
  __shared__ float s0[256];
  __shared__ float s1[256];
  long i = (long)blockIdx.x * 256 + threadIdx.x;
  int c = (int)((i / HW) % C);
  float v = x[i];
  s0[threadIdx.x] = v;
  s1[threadIdx.x] = v * v;
  __syncthreads();
  for (int off = 128; off > 0; off >>= 1) {
    if ((int)threadIdx.x < off) {
      s0[threadIdx.x] += s0[threadIdx.x + off];
      s1[threadIdx.x] += s1[threadIdx.x + off];
    }
    __syncthreads();
  }
  if (threadIdx.x == 0) {
    atomicAdd(&stats[2 * c + 0], s0[0]);
    atomicAdd(&stats[2 * c + 1], s1[0]);
  }
}

__global__ void k_bn_lrelu(float* __restrict__ x, const float* __restrict__ stats,
                           const float* __restrict__ g, const float* __restrict__ b,
                           int C, int HW, float invCount, long total) {
  long i = (long)blockIdx.x * blockDim.x + threadIdx.x;
  if (i >= total) return;
  int c = (int)((i / HW) % C);
  float mean = stats[2 * c + 0] * invCount;
  float var  = stats[2 * c + 1] * invCount - mean * mean;
  float sc = g[c] * rsqrtf(var + EPS);
  float y = sc * (x[i] - mean) + b[c];
  x[i] = (y >= 0.0f) ? y : SLOPE * y;
}

// --------------------------------------------------- codebook squared norms
__global__ void k_cbnorm(const float* __restrict__ cb, float* __restrict__ cbn) {
  int j = blockIdx.x * blockDim.x + threadIdx.x;
  if (j < 1024) {
    float s = 0.0f;
    for (int k = 0; k < 64; ++k) { float v = cb[j * 64 + k]; s = fmaf(v, v, s); }
    cbn[j] = s;
  }
}

// ---------------------------------------------------------------- VQ kernel
// pq: (16,64,64,64) NCHW viewed as z: (65536, 64); cb: (1024, 64).
// One block = 16 z-rows. 8 waves x 8 column-tiles each = 1024 codes.
// S = z @ cb^T via V_WMMA_F32_16X16X4_F32; argmin of ||c||^2 - 2*S.
static __device__ __forceinline__ v8f wmma_f32_4(v2f a, v2f b, v8f c) {
#if __has_builtin(__builtin_amdgcn_wmma_f32_16x16x4_f32)
  return __builtin_amdgcn_wmma_f32_16x16x4_f32(false, a, false, b, (short)0, c,
                                               false, false);
#else
  asm volatile("v_wmma_f32_16x16x4_f32 %0, %1, %2, %0"
               : "+v"(c) : "v"(a), "v"(b));
  return c;
#endif
}

__global__ void __launch_bounds__(256) k_vq(const float* __restrict__ pq,
                                            const float* __restrict__ cb,
                                            const float* __restrict__ cbn,
                                            float* __restrict__ zq,
                                            float* __restrict__ lossAcc) {
  __shared__ float zs[16 * 64];   // z tile [row][k]
  __shared__ float cd[2048];      // per-lane candidates: dist
  __shared__ int   ci[2048];      // per-lane candidates: idx
  __shared__ float pd[256];
  __shared__ int   pi[256];
  __shared__ int   rowIdx[16];
  __shared__ float lred[256];

  const int tid  = threadIdx.x;
  const int row0 = blockIdx.x * 16;     // base z row
  const int b    = row0 >> 12;          // / 4096
  const int hw0  = row0 & 4095;

  // stage z tile: zs[row*64+k] = pq[((b*64+k)<<12) + hw0 + row]
  for (int e = tid; e < 1024; e += 256) {
    int row = e >> 6, k = e & 63;
    zs[e] = pq[(((long)b * 64 + k) << 12) + hw0 + row];
  }
  __syncthreads();

  const int wave  = tid >> 5;
  const int lane  = tid & 31;
  const int lhalf = lane & 15;
  const int hi    = lane >> 4;          // 0: K=0,1  1: K=2,3 (A/B f32 layout)

  // A fragments: 16 K-steps of 4; A is 16x4 f32 (M = lhalf)
  v2f a[16];
  #pragma unroll
  for (int k0 = 0; k0 < 16; ++k0) {
    int base = lhalf * 64 + k0 * 4 + hi * 2;
    a[k0].x = zs[base];
    a[k0].y = zs[base + 1];
  }

  float best[8];
  int   bidx[8];
  #pragma unroll
  for (int r = 0; r < 8; ++r) { best[r] = 3.4e38f; bidx[r] = 0; }

  for (int t = 0; t < 8; ++t) {
    const int n0 = (wave * 8 + t) * 16;   // column tile base (ascending per wave)
    const int nc = n0 + lhalf;            // this lane's code index
    v8f acc = {};
    #pragma unroll
    for (int k0 = 0; k0 < 16; ++k0) {
      const float* bp = cb + (long)nc * 64 + k0 * 4 + hi * 2;
      v2f bf; bf.x = bp[0]; bf.y = bp[1];
      acc = wmma_f32_4(a[k0], bf, acc);
    }
    const float cn = cbn[nc];
    #pragma unroll
    for (int r = 0; r < 8; ++r) {         // acc[r]: row = r + 8*hi, col = nc
      float d = cn - 2.0f * acc[r];
      if (d < best[r]) { best[r] = d; bidx[r] = nc; }
    }
  }

  #pragma unroll
  for (int r = 0; r < 8; ++r) { cd[tid * 8 + r] = best[r]; ci[tid * 8 + r] = bidx[r]; }
  __syncthreads();

  // reduce across lanes/waves: row m held by lanes (m>>3)*16..+15, vgpr m&7
  {
    int m = tid >> 4, sub = tid & 15;
    int hi2 = m >> 3, r = m & 7;
    float bd = 3.4e38f; int bi = 0x7fffffff;
    for (int w = 0; w < 8; ++w) {
      int l = w * 32 + hi2 * 16 + sub;
      float d = cd[l * 8 + r]; int idv = ci[l * 8 + r];
      if (d < bd || (d == bd && idv < bi)) { bd = d; bi = idv; }
    }
    pd[tid] = bd; pi[tid] = bi;
  }
  __syncthreads();
  if (tid < 16) {
    float bd = 3.4e38f; int bi = 0x7fffffff;
    for (int s = 0; s < 16; ++s) {
      float d = pd[tid * 16 + s]; int idv = pi[tid * 16 + s];
      if (d < bd || (d == bd && idv < bi)) { bd = d; bi = idv; }
    }
    rowIdx[tid] = bi;
  }
  __syncthreads();

  // gather q = cb[idx], write zq (NCHW), accumulate sum (z-q)^2
  float ls = 0.0f;
  for (int e = tid; e < 1024; e += 256) {
    int row = e >> 6, c = e & 63;
    float qv = cb[(long)rowIdx[row] * 64 + c];
    zq[(((long)b * 64 + c) << 12) + hw0 + row] = qv;
    float diff = zs[e] - qv;
    ls = fmaf(diff, diff, ls);
  }
  lred[tid] = ls;
  __syncthreads();
  for (int off = 128; off > 0; off >>= 1) {
    if (tid < off) lred[tid] += lred[tid + off];
    __syncthreads();
  }
  if (tid == 0) atomicAdd(lossAcc, lred[0]);
}

__global__ void k_finalize_loss(const float* __restrict__ lossAcc,
                                float* __restrict__ outLoss) {
  // quantized = codebook + 0.25*commitment; both equal mean((z-q)^2) in fwd
  outLoss[0] = lossAcc[0] * 1.25f / (65536.0f * 64.0f);
}

// ------------------------------------------------------------------- driver
extern "C" void kernel_launch(void* const* d_in, const int* in_sizes, int n_in,
                              void* d_out, int out_size, void* d_ws, size_t ws_size,
                              hipStream_t stream) {
  const float* x      = (const float*)d_in[0];
  const float* cb     = (const float*)d_in[1];
  const float* e1_w   = (const float*)d_in[2];
  const float* e1_b   = (const float*)d_in[3];
  const float* bn1_g  = (const float*)d_in[4];
  const float* bn1_b  = (const float*)d_in[5];
  const float* e2_w   = (const float*)d_in[6];
  const float* e2_b   = (const float*)d_in[7];
  const float* bn2_g  = (const float*)d_in[8];
  const float* bn2_b  = (const float*)d_in[9];
  const float* preq_w = (const float*)d_in[10];
  const float* preq_b = (const float*)d_in[11];
  const float* postq_w= (const float*)d_in[12];
  const float* postq_b= (const float*)d_in[13];
  const float* d1_w   = (const float*)d_in[14];
  const float* d1_b   = (const float*)d_in[15];
  const float* bn3_g  = (const float*)d_in[16];
  const float* bn3_b  = (const float*)d_in[17];
  const float* d2_w   = (const float*)d_in[18];
  const float* d2_b   = (const float*)d_in[19];
  const float* bn4_g  = (const float*)d_in[20];
  const float* bn4_b  = (const float*)d_in[21];
  const float* d3_w   = (const float*)d_in[22];
  const float* d3_b   = (const float*)d_in[23];
  float* out = (float*)d_out;
  float* ws  = (float*)d_ws;

  // workspace layout (floats), with buffer reuse
  float* A     = ws;               // 4,194,304  h1 / pq / hd1
  float* Bb    = ws + 4194304;     //   262,144  h2 / hpost
  float* Cc    = ws + 4456448;     // 4,194,304  zq
  float* D     = ws + 8650752;     // 33,554,432 hd2
  float* stats = ws + 42205184;    // 64 (sum, sumsq interleaved per channel)
  float* cbn   = ws + 42205312;    // 1024
  float* loss  = ws + 42206336;    // 1

  const int T = 256;

  // ---- encoder conv1: (16,3,256,256) -> (16,16,128,128), stride 2
  k_conv2d<<<16384, T, 0, stream>>>(x, e1_w, e1_b, A, 16, 3, 256, 256, 16, 128, 128, 2, 0);
  k_zero<<<1, 64, 0, stream>>>(stats, 64);
  k_bn_stats<<<16384, T, 0, stream>>>(A, stats, 16, 16384);
  k_bn_lrelu<<<16384, T, 0, stream>>>(A, stats, bn1_g, bn1_b, 16, 16384,
                                      1.0f / 262144.0f, 4194304L);

  // ---- encoder conv2: (16,16,128,128) -> (16,4,64,64), stride 2
  k_conv2d<<<1024, T, 0, stream>>>(A, e2_w, e2_b, Bb, 16, 16, 128, 128, 4, 64, 64, 2, 0);
  k_zero<<<1, 64, 0, stream>>>(stats, 64);
  k_bn_stats<<<1024, T, 0, stream>>>(Bb, stats, 4, 4096);
  k_bn_lrelu<<<1024, T, 0, stream>>>(Bb, stats, bn2_g, bn2_b, 4, 4096,
                                     1.0f / 65536.0f, 262144L);

  // ---- pre-quant conv: (16,4,64,64) -> (16,64,64,64), stride 1
  k_conv2d<<<16384, T, 0, stream>>>(Bb, preq_w, preq_b, A, 16, 4, 64, 64, 64, 64, 64, 1, 0);

  // ---- vector quantization (WMMA GEMM + argmin + gather + loss)
  k_cbnorm<<<4, T, 0, stream>>>(cb, cbn);
  k_zero<<<1, 32, 0, stream>>>(loss, 1);
  k_vq<<<4096, T, 0, stream>>>(A, cb, cbn, Cc, loss);

  // ---- post-quant conv: (16,64,64,64) -> (16,4,64,64), stride 1
  k_conv2d<<<1024, T, 0, stream>>>(Cc, postq_w, postq_b, Bb, 16, 64, 64, 64, 4, 64, 64, 1, 0);

  // ---- decoder convT1: (16,4,64,64) -> (16,16,128,128)
  k_convT<<<16384, T, 0, stream>>>(Bb, d1_w, d1_b, A, 16, 4, 64, 64, 16);
  k_zero<<<1, 64, 0, stream>>>(stats, 64);
  k_bn_stats<<<16384, T, 0, stream>>>(A, stats, 16, 16384);
  k_bn_lrelu<<<16384, T, 0, stream>>>(A, stats, bn3_g, bn3_b, 16, 16384,
                                      1.0f / 262144.0f, 4194304L);

  // ---- decoder convT2: (16,16,128,128) -> (16,32,256,256)
  k_convT<<<131072, T, 0, stream>>>(A, d2_w, d2_b, D, 16, 16, 128, 128, 32);
  k_zero<<<1, 64, 0, stream>>>(stats, 64);
  k_bn_stats<<<131072, T, 0, stream>>>(D, stats, 32, 65536);
  k_bn_lrelu<<<131072, T, 0, stream>>>(D, stats, bn4_g, bn4_b, 32, 65536,
                                       1.0f / 1048576.0f, 33554432L);

  // ---- output conv + tanh: (16,32,256,256) -> (16,3,256,256)
  k_conv2d<<<12288, T, 0, stream>>>(D, d3_w, d3_b, out, 16, 32, 256, 256, 3, 256, 256, 1, 1);

  // ---- scalar loss output at d_out[3145728]
  k_finalize_loss<<<1, 1, 0, stream>>>(loss, out + 3145728);
}